// JointDGMRF_32581621907832
// MI455X (gfx1250) — compile-verified
//
#include <hip/hip_runtime.h>
#include <math.h>

typedef __attribute__((ext_vector_type(2))) float v2f;
typedef __attribute__((ext_vector_type(8))) float v8f;

#define TCH 16

// ---------------------------------------------------------------- utilities
__global__ void zero_kernel(int* __restrict__ p, long n) {
    long i = (long)blockIdx.x * blockDim.x + threadIdx.x;
    if (i < n) p[i] = 0;
}

__global__ void degree_kernel(const int* __restrict__ src, int* __restrict__ deg, int E) {
    int e = blockIdx.x * blockDim.x + threadIdx.x;
    if (e < E) atomicAdd(&deg[src[e]], 1);
}

__global__ void logdeg_kernel(const int* __restrict__ deg, float* __restrict__ logdeg, int N) {
    int n = blockIdx.x * blockDim.x + threadIdx.x;
    if (n < N) logdeg[n] = logf((float)deg[n]);   // deg==0 -> -inf, matches jnp.log(0)
}

// ------------------------------------------------- [T,N] -> [N,T] via WMMA
// One wave transposes one 16x16 f32 tile (16 nodes) using
// v_wmma_f32_16x16x4_f32 with an identity B matrix: D = A * I reproduces the
// tile, but the A-layout (row M = lane, K per VGPR) vs D-layout (row M per
// VGPR, col N = lane) difference performs the transpose in the matrix pipe.
// A and B share the (lane-half h, vgpr v) -> K = 2h + v mapping (ISA 7.12.2).
__global__ void transpose_in_wmma(const float* __restrict__ xin,
                                  float* __restrict__ xw, int N) {
    int wave = (blockIdx.x * blockDim.x + threadIdx.x) >> 5;
    int lane = threadIdx.x & 31;
    int n0 = wave << 4;                 // first node (column) of this tile
    if (n0 >= N) return;                // wave-uniform: EXEC stays all-ones
    int m = lane & 15;                  // A row (= channel t); also D col N
    int h = lane >> 4;                  // lane half
    const size_t rowBase = (size_t)m * (size_t)N;

    v8f c = {0.f, 0.f, 0.f, 0.f, 0.f, 0.f, 0.f, 0.f};
#pragma unroll
    for (int k = 0; k < 4; ++k) {
        int col = 4 * k + 2 * h;        // global K of this lane's a.x
        int j0 = n0 + col;
        int j1 = j0 + 1;
        if (j0 > N - 1) j0 = N - 1;     // clamped duplicate loads for tail;
        if (j1 > N - 1) j1 = N - 1;     // identity-B => col j of D depends only on col j of A
        v2f a;
        a.x = xin[rowBase + j0];
        a.y = xin[rowBase + j1];
        v2f b;
        b.x = (m == col)     ? 1.0f : 0.0f;   // B[K=4k+2h  , N=m]
        b.y = (m == col + 1) ? 1.0f : 0.0f;   // B[K=4k+2h+1, N=m]
        c = __builtin_amdgcn_wmma_f32_16x16x4_f32(false, a, false, b,
                                                  (short)0, c, false, false);
    }
    // D layout: lane holds D[h*8 + r, m], r = 0..7  ==  X_tile[t, m]
    int node = n0 + m;
    if (node < N) {
        float4* dst = (float4*)(xw + (size_t)node * TCH + h * 8);
        dst[0] = make_float4(c[0], c[1], c[2], c[3]);
        dst[1] = make_float4(c[4], c[5], c[6], c[7]);
    }
}

// ------------------------------------------------------------- edge scatter
__global__ void edge_kernel(const int* __restrict__ src, const int* __restrict__ dst,
                            const float* __restrict__ logdeg,
                            const float* __restrict__ xw, float* __restrict__ agg,
                            const float* __restrict__ gamma, int layer, int E) {
    int e = blockIdx.x * blockDim.x + threadIdx.x;
    if (e >= E) return;
    // stream-ahead prefetch of the edge-index arrays (global_prefetch_b8;
    // speculative prefetch is silently dropped past end of mapping)
    __builtin_prefetch(src + e + 8192, 0, 1);
    __builtin_prefetch(dst + e + 8192, 0, 1);

    float g  = gamma[layer];                      // uniform scalar load
    float dp = 1.0f / (1.0f + expf(-g));          // sigmoid
    int s = src[e];
    int d = dst[e];
    float ew = expf((dp - 1.0f) * logdeg[d]);     // deg[dst]^(dp-1)

    const float4* xr = (const float4*)(xw + (size_t)s * TCH);
    float4 a0 = xr[0], a1 = xr[1], a2 = xr[2], a3 = xr[3];
    float* ag = agg + (size_t)d * TCH;
    atomicAdd(ag + 0,  a0.x * ew);  atomicAdd(ag + 1,  a0.y * ew);
    atomicAdd(ag + 2,  a0.z * ew);  atomicAdd(ag + 3,  a0.w * ew);
    atomicAdd(ag + 4,  a1.x * ew);  atomicAdd(ag + 5,  a1.y * ew);
    atomicAdd(ag + 6,  a1.z * ew);  atomicAdd(ag + 7,  a1.w * ew);
    atomicAdd(ag + 8,  a2.x * ew);  atomicAdd(ag + 9,  a2.y * ew);
    atomicAdd(ag + 10, a2.z * ew);  atomicAdd(ag + 11, a2.w * ew);
    atomicAdd(ag + 12, a3.x * ew);  atomicAdd(ag + 13, a3.y * ew);
    atomicAdd(ag + 14, a3.z * ew);  atomicAdd(ag + 15, a3.w * ew);
}

// ------------------------------------- layer-0 combine (in place) + agg clear
__global__ void combine_mid_kernel(float* __restrict__ xw, float* __restrict__ agg,
                                   const float* __restrict__ logdeg,
                                   const float* __restrict__ alpha1,
                                   const float* __restrict__ gamma,
                                   const float* __restrict__ bias,
                                   int layer, int N) {
    int n = blockIdx.x * blockDim.x + threadIdx.x;
    if (n >= N) return;
    float a1 = alpha1[layer];
    float sw = expf(a1);
    float nw = sw * tanhf(a1);
    float dp = 1.0f / (1.0f + expf(-gamma[layer]));
    float b  = bias[layer];
    float wr = sw * expf(dp * logdeg[n]);          // sw * deg^dp

    float4* xr = (float4*)(xw  + (size_t)n * TCH);
    float4* ag = (float4*)(agg + (size_t)n * TCH);
    const float4 z = make_float4(0.f, 0.f, 0.f, 0.f);
#pragma unroll
    for (int i = 0; i < 4; ++i) {
        float4 xv = xr[i];
        float4 av = ag[i];
        xv.x = xv.x * wr + nw * av.x + b;
        xv.y = xv.y * wr + nw * av.y + b;
        xv.z = xv.z * wr + nw * av.z + b;
        xv.w = xv.w * wr + nw * av.w + b;
        xr[i] = xv;
        ag[i] = z;                                  // re-zero agg for next layer
    }
}

// ---------------------- layer-1 combine, fused transpose back to [T,N] output
__global__ void combine_out_kernel(const float* __restrict__ xw,
                                   const float* __restrict__ agg,
                                   const float* __restrict__ logdeg,
                                   const float* __restrict__ alpha1,
                                   const float* __restrict__ gamma,
                                   const float* __restrict__ bias,
                                   int layer, int N, float* __restrict__ out) {
    int n = blockIdx.x * blockDim.x + threadIdx.x;
    if (n >= N) return;
    float a1 = alpha1[layer];
    float sw = expf(a1);
    float nw = sw * tanhf(a1);
    float dp = 1.0f / (1.0f + expf(-gamma[layer]));
    float b  = bias[layer];
    float wr = sw * expf(dp * logdeg[n]);

    const float* xr = xw  + (size_t)n * TCH;
    const float* ag = agg + (size_t)n * TCH;
#pragma unroll
    for (int t = 0; t < TCH; ++t) {
        // fixed t, consecutive lanes -> consecutive n : coalesced 128B stores
        out[(size_t)t * N + n] = xr[t] * wr + nw * ag[t] + b;
    }
}

// --------------------------------------------------------------------- host
extern "C" void kernel_launch(void* const* d_in, const int* in_sizes, int n_in,
                              void* d_out, int out_size, void* d_ws, size_t ws_size,
                              hipStream_t stream) {
    const float* x      = (const float*)d_in[0];   // [T, N]
    const int*   ei     = (const int*)d_in[1];     // [2, E]
    const float* alpha1 = (const float*)d_in[2];   // [L]
    // d_in[3] = alpha2 : unused (reference uses alpha1 for neighbor weight)
    const float* gamma  = (const float*)d_in[4];   // [L]
    const float* bias   = (const float*)d_in[5];   // [L]

    const int N = in_sizes[0] / TCH;
    const int E = in_sizes[1] / 2;
    const int* src = ei;
    const int* dst = ei + E;

    // workspace layout: [agg N*16 f][deg N i][logdeg N f][xw N*16 f]  (~13.6 MB)
    char* ws = (char*)d_ws;
    size_t aggOff = 0;
    size_t degOff = aggOff + (size_t)N * TCH * sizeof(float);
    size_t ldOff  = degOff + (size_t)N * sizeof(int);
    size_t xwOff  = (ldOff + (size_t)N * sizeof(float) + 15) & ~(size_t)15;
    float* agg    = (float*)(ws + aggOff);
    int*   deg    = (int*)  (ws + degOff);
    float* logdeg = (float*)(ws + ldOff);
    float* xw     = (float*)(ws + xwOff);

    const int TB = 256;
    long zn = (long)N * TCH + N;                     // agg + deg are contiguous
    zero_kernel<<<(int)((zn + TB - 1) / TB), TB, 0, stream>>>((int*)ws, zn);

    degree_kernel<<<(E + TB - 1) / TB, TB, 0, stream>>>(src, deg, E);
    logdeg_kernel<<<(N + TB - 1) / TB, TB, 0, stream>>>(deg, logdeg, N);

    int tiles = (N + 15) / 16;                       // one 16-node tile per wave
    transpose_in_wmma<<<(tiles + 7) / 8, TB, 0, stream>>>(x, xw, N);

    // layer 0
    edge_kernel<<<(E + TB - 1) / TB, TB, 0, stream>>>(src, dst, logdeg, xw, agg, gamma, 0, E);
    combine_mid_kernel<<<(N + TB - 1) / TB, TB, 0, stream>>>(xw, agg, logdeg,
                                                             alpha1, gamma, bias, 0, N);
    // layer 1
    edge_kernel<<<(E + TB - 1) / TB, TB, 0, stream>>>(src, dst, logdeg, xw, agg, gamma, 1, E);
    combine_out_kernel<<<(N + TB - 1) / TB, TB, 0, stream>>>(xw, agg, logdeg,
                                                             alpha1, gamma, bias, 1, N,
                                                             (float*)d_out);
}